// Haar3D_88270167867694
// MI455X (gfx1250) — compile-verified
//
#include <hip/hip_runtime.h>
#include <hip/hip_bf16.h>

// Haar 3D (2x2x2 stride-2, 8-filter bank) as a streaming WMMA kernel.
//
// Formulated as D = X * H^T with two chained V_WMMA_F32_16X16X4_F32
// (K = 0..3 then 4..7), f32 end-to-end:
//   A (16x4)  = 16 consecutive-ow blocks; lane L vgpr v -> block L%16,
//               element k = 2*(L/16)+v  ==> exactly one b64 global load/lane.
//   B (4x16)  = H^T, B[k,o] = 0.125*(-1)^popc(o&k), cols o>=8 zero
//               (constant, built per-lane in registers).
//   D         = vgpr r holds block r (+8 for hi lanes), filter o = lane%16
//               ==> 8 consecutive ow per lane ==> two b128 NT stores.
//
// Roofline: 256MB in + 256MB out, ~2 GFLOP -> pure HBM bound (~22us @
// 23.3 TB/s). Each wave loops 16 oh-rows so prologue + Haar-constant setup
// is amortized: ~14 instructions per 1KB of traffic -> issue time well
// below the bandwidth limit. NT hints since 512MB streams with zero reuse
// (exceeds the 192MB L2).

typedef __attribute__((ext_vector_type(2))) float v2f;
typedef __attribute__((ext_vector_type(4))) float v4f;
typedef __attribute__((ext_vector_type(8))) float v8f;

#define IN_W     128
#define IN_PLANE (128LL * 128LL)
#define IN_CHAN  (128LL * 128LL * 128LL)
#define OUT_W    64
#define OUT_PLANE (64LL * 64LL)
#define OUT_CHAN  (64LL * 64LL * 64LL)

__global__ __launch_bounds__(256) void haar3d_wmma_kernel(
    const float* __restrict__ x, float* __restrict__ out) {
  const int lane = threadIdx.x & 31;
  const int wv = threadIdx.x >> 5;

  // task = (nc, od, ohg, owg): 32 * 64 * 4 * 4 = 32768 waves, 16 oh each
  const int g   = blockIdx.x * 8 + wv;
  const int owg = g & 3;
  const int ohg = (g >> 2) & 3;
  const int od  = (g >> 4) & 63;
  const int nc  = g >> 10;                 // n*16 + c, 0..31

  const int half = lane >> 4;              // K-row pair select
  const int col  = lane & 15;

  // ---- constant B = H^T (cols o>=8 zero-padded) ----
  v2f bh1, bh2;
#pragma unroll
  for (int v = 0; v < 2; ++v) {
    const int k1 = 2 * half + v;           // step-1 block element
    const int k2 = k1 + 4;                 // step-2 block element
    const float s1 = (__popc(col & k1) & 1) ? -0.125f : 0.125f;
    const float s2 = (__popc(col & k2) & 1) ? -0.125f : 0.125f;
    bh1[v] = (col < 8) ? s1 : 0.0f;
    bh2[v] = (col < 8) ? s2 : 0.0f;
  }

  // ---- A loads: lane L = block col, rows (d=0/1, h=half, w=0..1) ----
  const float* p = x + (long long)nc * IN_CHAN
                     + (long long)(2 * od) * IN_PLANE
                     + (long long)(2 * (ohg * 16) + half) * IN_W
                     + 2 * (owg * 16 + col);

  // ---- store mapping: lanes with (lane%16) < 8 own channel o ----
  const bool active = (lane & 8) == 0;
  const int o  = lane & 7;
  const int n0 = (lane >> 4) * 8;          // ow sub-offset for hi lanes
  float* yo = out + (long long)(nc * 8 + o) * OUT_CHAN
                  + (long long)od * OUT_PLANE
                  + (long long)(ohg * 16) * OUT_W
                  + owg * 16 + n0;

#pragma unroll 4
  for (int i = 0; i < 16; ++i) {
    v2f a1 = __builtin_nontemporal_load((const v2f*)p);              // d=0
    v2f a2 = __builtin_nontemporal_load((const v2f*)(p + IN_PLANE)); // d=1
    v8f d = {};
    d = __builtin_amdgcn_wmma_f32_16x16x4_f32(
        /*neg_a=*/false, a1, /*neg_b=*/false, bh1,
        /*c_mod=*/(short)0, d, /*reuse_a=*/false, /*reuse_b=*/false);
    d = __builtin_amdgcn_wmma_f32_16x16x4_f32(
        false, a2, false, bh2, (short)0, d, false, false);
    if (active) {
      v4f lo = __builtin_shufflevector(d, d, 0, 1, 2, 3);
      v4f hi = __builtin_shufflevector(d, d, 4, 5, 6, 7);
      __builtin_nontemporal_store(lo, (v4f*)yo);
      __builtin_nontemporal_store(hi, (v4f*)(yo + 4));
    }
    p  += 2 * IN_W;    // next oh row (ih += 2)
    yo += OUT_W;
  }
}

extern "C" void kernel_launch(void* const* d_in, const int* in_sizes, int n_in,
                              void* d_out, int out_size, void* d_ws, size_t ws_size,
                              hipStream_t stream) {
  (void)in_sizes; (void)n_in; (void)d_ws; (void)ws_size; (void)out_size;
  const float* x = (const float*)d_in[0];
  // d_in[1] (Haar bank) is a fixed constant reproduced analytically in-kernel.
  float* out = (float*)d_out;

  // 32768 waves / 8 waves-per-block = 4096 blocks
  haar3d_wmma_kernel<<<4096, 256, 0, stream>>>(x, out);
}